// GCNModel_71365176590644
// MI455X (gfx1250) — compile-verified
//
#include <hip/hip_runtime.h>

typedef float v2f __attribute__((ext_vector_type(2)));
typedef float v8f __attribute__((ext_vector_type(8)));

// ---------------------------------------------------------------------------
// Degree / normalization kernels
// ---------------------------------------------------------------------------
__global__ __launch_bounds__(256) void deg_init_kernel(float* deg, int N) {
    int i = blockIdx.x * blockDim.x + threadIdx.x;
    if (i < N) deg[i] = 1.0f;  // self-loop contributes 1 to every node's degree
}

__global__ __launch_bounds__(256) void deg_edges_kernel(const int* __restrict__ dst,
                                                        float* deg, int E) {
    int e = blockIdx.x * blockDim.x + threadIdx.x;
    if (e < E) atomicAdd(&deg[dst[e]], 1.0f);
}

__global__ __launch_bounds__(256) void deg_rsqrt_kernel(const float* __restrict__ deg,
                                                        float* dinv, int N) {
    int i = blockIdx.x * blockDim.x + threadIdx.x;
    if (i < N) {
        float d = deg[i];
        dinv[i] = (d > 0.0f) ? rsqrtf(d) : 0.0f;
    }
}

// ---------------------------------------------------------------------------
// WMMA fp32 GEMM:  Out[M,Nn] = A[M,K] @ B[K,Nn] (+bias) (+relu)
//
// Block = 8 waves = 128(M) x 64(N) macro-tile; each wave owns a 16x64 strip
// (4 accumulators). B is double-buffer staged through LDS in 32x64 chunks.
//
// LDS layout is K-pair interleaved: element (k,n) lives at
//   sB[buf][(k>>1)*LDP + 2n + (k&1)],  LDP = 160
// so a B fragment {B[ka][n], B[ka+1][n]} is ONE contiguous 8-byte read ->
// one ds_load_b64 straight into an even-aligned VGPR pair (no v_mov fixups).
// Pairrow stride 160 = 32 mod 64 banks => lanes 0-15 (pairrow 2j) and lanes
// 16-31 (pairrow 2j+1) read complementary 32-bank halves: conflict-free.
//
// Pipeline: next chunk's global loads are issued before the current chunk's
// 32 WMMAs, kept in registers, stored to the idle buffer after compute;
// one barrier per chunk. A fragments for a whole chunk are pre-loaded as a
// clause of 8 b64 global loads. V_WMMA_F32_16X16X4_F32 is exact fp32 math.
// Requires: M % 16 == 0, K % 32 == 0, Nn % 64 == 0 (all satisfied here).
// ---------------------------------------------------------------------------
template <bool BIAS, bool RELU>
__global__ __launch_bounds__(256) void gemm_wmma_f32(const float* __restrict__ A,
                                                     const float* __restrict__ B,
                                                     const float* __restrict__ bias,
                                                     float* __restrict__ Out,
                                                     int M, int K, int Nn) {
    constexpr int BK  = 32;    // K-chunk staged per iteration
    constexpr int LDP = 160;   // padded LDS pair-row stride (floats)
    __shared__ float sB[2][(BK / 2) * LDP];

    const int tid   = threadIdx.x;
    const int lane  = tid & 31;
    const int wave  = tid >> 5;
    const int strips_n = Nn >> 6;             // 64-wide N strips
    const int tiles_m  = M >> 4;
    const int mb   = blockIdx.x / strips_n;   // 128-row macro block
    const int ns   = blockIdx.x - mb * strips_n;
    const int tile_m = mb * 8 + wave;
    const bool valid = (tile_m < tiles_m);    // wave-uniform
    const int nbase  = ns * 64;

    const int half = lane >> 4;               // 0: K pair {0,1}, 1: {2,3}
    const int l16  = lane & 15;
    const float* __restrict__ Arow = A + (size_t)(tile_m * 16 + l16) * K;

    // Cooperative B staging: thread owns pair-row p = tid/16, 4 cols c0.
    // Loads float4 from rows 2p and 2p+1, writes 4 interleaved b64 stores.
    const int p  = tid >> 4;                  // 0..15
    const int c0 = (tid & 15) * 4;            // 0,4,...,60
    const float* __restrict__ Bg0 = B + (size_t)(2 * p)     * Nn + nbase + c0;
    const float* __restrict__ Bg1 = B + (size_t)(2 * p + 1) * Nn + nbase + c0;
    float* const sDstA = &sB[0][p * LDP + 2 * c0];
    float* const sDstB = &sB[1][p * LDP + 2 * c0];

    const int chunks = K / BK;

    // Preload chunk 0 into buffer 0
    {
        const float4 r0 = *reinterpret_cast<const float4*>(Bg0);
        const float4 r1 = *reinterpret_cast<const float4*>(Bg1);
        reinterpret_cast<v2f*>(sDstA)[0] = v2f{r0.x, r1.x};
        reinterpret_cast<v2f*>(sDstA)[1] = v2f{r0.y, r1.y};
        reinterpret_cast<v2f*>(sDstA)[2] = v2f{r0.z, r1.z};
        reinterpret_cast<v2f*>(sDstA)[3] = v2f{r0.w, r1.w};
    }
    __syncthreads();

    v8f c0acc = {}, c1acc = {}, c2acc = {}, c3acc = {};
    for (int c = 0; c < chunks; ++c) {
        // Issue next chunk's global loads first (latency overlapped w/ WMMAs)
        float4 r0, r1;
        const bool has_next = (c + 1 < chunks);
        if (has_next) {
            const size_t adv = (size_t)(c + 1) * BK * Nn;
            r0 = *reinterpret_cast<const float4*>(Bg0 + adv);
            r1 = *reinterpret_cast<const float4*>(Bg1 + adv);
        }

        if (valid) {
            const float* __restrict__ Ac = Arow + c * BK;
            v2f af[8];
#pragma unroll
            for (int j = 0; j < 8; ++j)
                af[j] = *reinterpret_cast<const v2f*>(Ac + j * 4 + half * 2);

            const float* __restrict__ sb = sB[c & 1];
#pragma unroll
            for (int j = 0; j < 8; ++j) {
                // fragment t: one contiguous b64 read per lane
                const int base = (j * 2 + half) * LDP + 2 * l16;
                const v2f b0 = *reinterpret_cast<const v2f*>(sb + base);
                const v2f b1 = *reinterpret_cast<const v2f*>(sb + base + 32);
                const v2f b2 = *reinterpret_cast<const v2f*>(sb + base + 64);
                const v2f b3 = *reinterpret_cast<const v2f*>(sb + base + 96);
                c0acc = __builtin_amdgcn_wmma_f32_16x16x4_f32(false, af[j], false, b0, (short)0, c0acc, false, false);
                c1acc = __builtin_amdgcn_wmma_f32_16x16x4_f32(false, af[j], false, b1, (short)0, c1acc, false, false);
                c2acc = __builtin_amdgcn_wmma_f32_16x16x4_f32(false, af[j], false, b2, (short)0, c2acc, false, false);
                c3acc = __builtin_amdgcn_wmma_f32_16x16x4_f32(false, af[j], false, b3, (short)0, c3acc, false, false);
            }
        }

        // Store next chunk into the idle buffer (its last readers finished
        // before the barrier that ended iteration c-1), then sync once.
        if (has_next) {
            float* const d = (c & 1) ? sDstA : sDstB;
            reinterpret_cast<v2f*>(d)[0] = v2f{r0.x, r1.x};
            reinterpret_cast<v2f*>(d)[1] = v2f{r0.y, r1.y};
            reinterpret_cast<v2f*>(d)[2] = v2f{r0.z, r1.z};
            reinterpret_cast<v2f*>(d)[3] = v2f{r0.w, r1.w};
        }
        __syncthreads();
    }

    if (!valid) return;

    v8f acc[4] = {c0acc, c1acc, c2acc, c3acc};
#pragma unroll
    for (int t = 0; t < 4; ++t) {
        const int n  = nbase + t * 16 + l16;
        const float bv = BIAS ? bias[n] : 0.0f;
#pragma unroll
        for (int r = 0; r < 8; ++r) {
            const int row = tile_m * 16 + r + half * 8;
            float v = acc[t][r] + bv;
            if (RELU) v = v > 0.0f ? v : 0.0f;
            Out[(size_t)row * Nn + n] = v;
        }
    }
}

// ---------------------------------------------------------------------------
// GCN aggregation.
// agg_init fuses the self-loop term (dinv[i]^2 * hw[i]) and the output bias.
// agg_edges: one wave per edge; each lane owns 8 contiguous features so the
// gather is two global_load_b128 per lane (wave covers one contiguous 1KB
// row); scatter is global_atomic_add_f32 per element.
// ---------------------------------------------------------------------------
__global__ __launch_bounds__(256) void agg_init_kernel(const float* __restrict__ hw,
                                                       const float* __restrict__ dinv,
                                                       const float* __restrict__ bias,
                                                       float* __restrict__ agg,
                                                       int F) {
    const int i = blockIdx.x;        // node
    const int f = threadIdx.x;       // feature (F == blockDim.x == 256)
    const float di = dinv[i];
    agg[(size_t)i * F + f] = hw[(size_t)i * F + f] * di * di + bias[f];
}

__global__ __launch_bounds__(256) void agg_edges_kernel(const int* __restrict__ src,
                                                        const int* __restrict__ dst,
                                                        const float* __restrict__ hw,
                                                        const float* __restrict__ dinv,
                                                        float* __restrict__ agg,
                                                        int E, int F) {
    const int lane = threadIdx.x & 31;
    const int wave = threadIdx.x >> 5;
    const int e = blockIdx.x * 8 + wave;
    if (e >= E) return;
    const int s = src[e];
    const int d = dst[e];
    const float norm = dinv[s] * dinv[d];
    const float* __restrict__ hs = hw + (size_t)s * F;
    float* __restrict__ ad = agg + (size_t)d * F;
    // lane owns features [lane*8, lane*8+8) (+ stride for F > 256)
    for (int f0 = lane * 8; f0 < F; f0 += 32 * 8) {
        const float4 v0 = *reinterpret_cast<const float4*>(hs + f0);
        const float4 v1 = *reinterpret_cast<const float4*>(hs + f0 + 4);
        atomicAdd(&ad[f0 + 0], v0.x * norm);
        atomicAdd(&ad[f0 + 1], v0.y * norm);
        atomicAdd(&ad[f0 + 2], v0.z * norm);
        atomicAdd(&ad[f0 + 3], v0.w * norm);
        atomicAdd(&ad[f0 + 4], v1.x * norm);
        atomicAdd(&ad[f0 + 5], v1.y * norm);
        atomicAdd(&ad[f0 + 6], v1.z * norm);
        atomicAdd(&ad[f0 + 7], v1.w * norm);
    }
}

// ---------------------------------------------------------------------------
// Tiny classifier: logits[B,C] = h4[B,oDim] @ Wc[oDim,C] + bc
// ---------------------------------------------------------------------------
__global__ __launch_bounds__(256) void classifier_kernel(const float* __restrict__ h4,
                                                         const float* __restrict__ Wc,
                                                         const float* __restrict__ bc,
                                                         float* __restrict__ out,
                                                         int B, int oDim, int C) {
    int t = blockIdx.x * blockDim.x + threadIdx.x;
    if (t >= B * C) return;
    int b = t / C;
    int c = t - b * C;
    const float* __restrict__ hr = h4 + (size_t)b * oDim;
    float acc = bc[c];
    for (int k = 0; k < oDim; ++k) acc += hr[k] * Wc[k * C + c];
    out[t] = acc;
}

// ---------------------------------------------------------------------------
// Launch
// ---------------------------------------------------------------------------
extern "C" void kernel_launch(void* const* d_in, const int* in_sizes, int n_in,
                              void* d_out, int out_size, void* d_ws, size_t ws_size,
                              hipStream_t stream) {
    const float* x   = (const float*)d_in[0];
    const int*   ei  = (const int*)d_in[1];
    const float* W1  = (const float*)d_in[2];
    const float* b1  = (const float*)d_in[3];
    const float* Wg1 = (const float*)d_in[4];
    const float* bg1 = (const float*)d_in[5];
    const float* Wg2 = (const float*)d_in[6];
    const float* bg2 = (const float*)d_in[7];
    const float* W2  = (const float*)d_in[8];
    const float* b2  = (const float*)d_in[9];
    const float* Wc  = (const float*)d_in[10];
    const float* bc  = (const float*)d_in[11];

    const int hid   = in_sizes[3];            // 256
    const int inDim = in_sizes[2] / hid;      // 768
    const int N     = in_sizes[0] / inDim;    // 50000
    const int E     = in_sizes[1] / 2;        // 800000
    const int oDim  = in_sizes[9];            // 128
    const int C     = in_sizes[11];           // 2
    const int B     = 1024;                   // batch_size (python scalar in setup)

    const int* src = ei;
    const int* dst = ei + E;

    // Workspace carve-up (floats)
    float* deg  = (float*)d_ws;                 // N
    float* dinv = deg + N;                      // N
    float* h1   = dinv + N;                     // N x hid   (reused later for h4)
    float* hw   = h1 + (size_t)N * hid;         // N x hid
    float* agg  = hw + (size_t)N * hid;         // N x hid
    float* h4   = h1;                           // B x oDim (h1 is dead by then)

    const int thr = 256;

    // --- degrees (shared by both convs) ---
    deg_init_kernel<<<(N + thr - 1) / thr, thr, 0, stream>>>(deg, N);
    deg_edges_kernel<<<(E + thr - 1) / thr, thr, 0, stream>>>(dst, deg, E);
    deg_rsqrt_kernel<<<(N + thr - 1) / thr, thr, 0, stream>>>(deg, dinv, N);

    auto gemm_grid = [](int M, int Nn) {
        int mblocks = (M / 16 + 7) / 8;       // 128-row macro blocks
        return mblocks * (Nn / 64);
    };

    // --- stage 1: h1 = relu(x @ W1 + b1) ---
    gemm_wmma_f32<true, true><<<gemm_grid(N, hid), thr, 0, stream>>>(
        x, W1, b1, h1, N, inDim, hid);

    // --- conv 1: hw = h1 @ Wg1 ; agg = scatter(hw) + bg1 ---
    gemm_wmma_f32<false, false><<<gemm_grid(N, hid), thr, 0, stream>>>(
        h1, Wg1, nullptr, hw, N, hid, hid);
    agg_init_kernel<<<N, hid, 0, stream>>>(hw, dinv, bg1, agg, hid);
    agg_edges_kernel<<<(E + 7) / 8, thr, 0, stream>>>(src, dst, hw, dinv, agg, E, hid);

    // --- conv 2: hw = agg @ Wg2 ; agg = scatter(hw) + bg2 (in place over old h2) ---
    gemm_wmma_f32<false, false><<<gemm_grid(N, hid), thr, 0, stream>>>(
        agg, Wg2, nullptr, hw, N, hid, hid);
    agg_init_kernel<<<N, hid, 0, stream>>>(hw, dinv, bg2, agg, hid);
    agg_edges_kernel<<<(E + 7) / 8, thr, 0, stream>>>(src, dst, hw, dinv, agg, E, hid);

    // --- stage 4: h4 = relu(agg[:B] @ W2 + b2) — only first B rows are needed ---
    gemm_wmma_f32<true, true><<<gemm_grid(B, oDim), thr, 0, stream>>>(
        agg, W2, b2, h4, B, hid, oDim);

    // --- stage 5: logits = h4 @ Wc + bc ---
    classifier_kernel<<<(B * C + thr - 1) / thr, thr, 0, stream>>>(
        h4, Wc, bc, (float*)d_out, B, oDim, C);
}